// BellmanFordLayerModified_22754736734767
// MI455X (gfx1250) — compile-verified
//
#include <hip/hip_runtime.h>
#include <stdint.h>

// Problem sizes (fixed by the reference)
#define V      1024     // nodes
#define NF     256      // embedding features
#define OUTW   257      // features + distance column
#define NBLK   32       // persistent workgroups
#define WGSIZE 1024     // threads per workgroup (32 wave32 waves)
#define WPB    32       // waves per block
#define DPB    32       // destination nodes per block
#define MAXIT  (V - 1)  // reference runs V-1 relaxation steps
#define INF    __builtin_huge_valf()

typedef unsigned int u32x4 __attribute__((ext_vector_type(4)));
typedef int          i32x4 __attribute__((ext_vector_type(4)));
typedef int          i32x8 __attribute__((ext_vector_type(8)));

#if __has_builtin(__builtin_amdgcn_tensor_load_to_lds)
#define HAVE_TDM 1
#else
#define HAVE_TDM 0
#endif

#if __has_builtin(__builtin_amdgcn_global_load_async_to_lds_b128)
#define HAVE_ASYNC_LDS 1
#else
#define HAVE_ASYNC_LDS 0
#endif

// Fail loudly on the DEVICE pass if the TDM builtin is missing; the host pass
// of hipcc never sees AMDGCN builtins, so it must skip this check.
#if defined(__HIP_DEVICE_COMPILE__)
static_assert(HAVE_TDM, "CDNA5 TDM builtin __builtin_amdgcn_tensor_load_to_lds not available");
#endif

__device__ __forceinline__ void wait_asynccnt0() {
#if __has_builtin(__builtin_amdgcn_s_wait_asynccnt)
    __builtin_amdgcn_s_wait_asynccnt(0);
#else
    asm volatile("s_wait_asynccnt 0" ::: "memory");
#endif
}

// ---------------------------------------------------------------------------
// Software grid barrier across the NBLK persistent workgroups.
// counter/gen live in d_ws and are re-initialized by init_kernel every call,
// so kernel_launch stays deterministic and leaves no live state behind.
// ---------------------------------------------------------------------------
__device__ __forceinline__ void grid_barrier(int* cnt, int* gen) {
    __syncthreads();
    if (threadIdx.x == 0) {
        __threadfence();  // publish this block's global writes (device scope)
        int g = __hip_atomic_load(gen, __ATOMIC_RELAXED, __HIP_MEMORY_SCOPE_AGENT);
        int t = __hip_atomic_fetch_add(cnt, 1, __ATOMIC_ACQ_REL, __HIP_MEMORY_SCOPE_AGENT);
        if (t == NBLK - 1) {
            __hip_atomic_store(cnt, 0, __ATOMIC_RELAXED, __HIP_MEMORY_SCOPE_AGENT);
            __hip_atomic_fetch_add(gen, 1, __ATOMIC_RELEASE, __HIP_MEMORY_SCOPE_AGENT);
        } else {
            while (__hip_atomic_load(gen, __ATOMIC_ACQUIRE, __HIP_MEMORY_SCOPE_AGENT) == g) {
                __builtin_amdgcn_s_sleep(2);
            }
        }
    }
    __syncthreads();
}

// ---------------------------------------------------------------------------
// TDM: load a 2-D tile (DPB contiguous columns x V rows, row stride V floats)
// of the adjacency matrix into LDS with one tensor_load_to_lds.
// D# group0: count=1, lds_addr, 57-bit global_addr, type=2.
// D# group1: data_size=4B, tensor_dim0/1 = 1024, tile_dim0=32, tile_dim1=1024,
//            tensor_dim0_stride = 1024 elements. Groups 2/3 zero (2-D tile).
// ---------------------------------------------------------------------------
__device__ __forceinline__ void tdm_load_adj_tile(const float* gsrc, unsigned lds_byte_addr) {
#if HAVE_TDM
    uint64_t ga = (uint64_t)(uintptr_t)gsrc;
    u32x4 g0;
    g0[0] = 1u;                                                  // count=1, user mode
    g0[1] = lds_byte_addr;                                       // LDS dest (bytes)
    g0[2] = (unsigned)(ga & 0xFFFFFFFFu);                        // global_addr[31:0]
    g0[3] = (unsigned)((ga >> 32) & 0x01FFFFFFu) | (2u << 30);   // addr[56:32] | type=2
    i32x8 g1;
    g1[0] = (int)(2u << 16);            // workgroup_mask=0, data_size=2 (4 bytes)
    g1[1] = (int)(1024u << 16);         // atomic_barrier_addr=0, tensor_dim0.lo16=1024
    g1[2] = (int)(1024u << 16);         // tensor_dim0.hi16=0,   tensor_dim1.lo16=1024
    g1[3] = (int)((unsigned)DPB << 16); // tensor_dim1.hi16=0,   tile_dim0=32
    g1[4] = (int)V;                     // tile_dim1=1024, tile_dim2=0
    g1[5] = (int)V;                     // tensor_dim0_stride.lo32 = 1024
    g1[6] = 0;                          // stride.hi16 = 0, tensor_dim1_stride.lo16 = 0
    g1[7] = 0;
    i32x4 gz = {0, 0, 0, 0};
#if __clang_major__ >= 23
    i32x8 gz8 = {0, 0, 0, 0, 0, 0, 0, 0};
    __builtin_amdgcn_tensor_load_to_lds(g0, g1, gz, gz, gz8, 0);
#else
    __builtin_amdgcn_tensor_load_to_lds(g0, g1, gz, gz, 0);
#endif
    __builtin_amdgcn_s_wait_tensorcnt(0);
#else
    (void)gsrc; (void)lds_byte_addr;   // host-pass parse only; device asserts HAVE_TDM
#endif
}

// ---------------------------------------------------------------------------
// Async staging of the 4 KB distance vector: global -> LDS without VGPRs.
// 256 lanes x b128 covers V floats; tracked by ASYNCcnt.
// The builtin's pointee type is int __vector(4); Sema adapts address spaces.
// ---------------------------------------------------------------------------
__device__ __forceinline__ void stage_d_to_lds(const float* dc, float* lds_d, int t) {
#if HAVE_ASYNC_LDS
    if (t < V / 4) {
        __builtin_amdgcn_global_load_async_to_lds_b128(
            (i32x4*)(dc + 4 * t),
            (i32x4*)(lds_d + 4 * t),
            0, 0);
        wait_asynccnt0();
    }
#else
    lds_d[t] = dc[t];
#endif
}

// ---------------------------------------------------------------------------
// Init: d0, convergence flags, barrier state.
// ---------------------------------------------------------------------------
__global__ void bf_init_kernel(float* dcur, float* dnext, int* flags, int* bar,
                               const int* srcp) {
    int t = threadIdx.x;
    int src = *srcp;
    dcur[t]  = (t == src) ? 0.0f : INF;
    dnext[t] = INF;
    flags[t] = 0;       // one flag per iteration, V >= MAXIT
    if (t < 2) bar[t] = 0;
}

// ---------------------------------------------------------------------------
// Persistent Bellman-Ford + epilogue.
// Block b owns dst in [32b, 32b+32). Wave w, lane l: sources [32w,32w+32),
// dst column l. adj slice lives in LDS for the whole kernel.
// ---------------------------------------------------------------------------
__global__ void __launch_bounds__(WGSIZE, 1)
bf_persistent_kernel(const float* __restrict__ adj,
                     const float* __restrict__ emb,
                     const int*   __restrict__ srcp,
                     float*       __restrict__ out,
                     float* dcur, float* dnext, int* flags, int* bar) {
    extern __shared__ float smem[];
    float* lds_adj  = smem;                 // V * DPB  floats (128 KB)
    float* lds_d    = lds_adj + V * DPB;    // V        floats (4 KB)
    float* lds_part = lds_d + V;            // WPB*DPB  floats (4 KB)

    const int t       = threadIdx.x;
    const int w       = t >> 5;             // wave id (wave32)
    const int l       = t & 31;             // lane id
    const int dstBase = blockIdx.x * DPB;

    // ---- Stage this block's adjacency slice into LDS once, via the TDM ----
    if (t < 32) {   // one wave issues the TDM op; EXEC ignored by TDM
        tdm_load_adj_tile(adj + dstBase, (unsigned)(uintptr_t)lds_adj);
    }
    __syncthreads();
    // Mask diagonal (s == dst) to +inf, matching the reference's eye-mask.
    if (t < DPB) lds_adj[(dstBase + t) * DPB + t] = INF;
    __syncthreads();

    // ---- Relaxation loop (Jacobi, double buffered, exact early exit) ----
    float* dc = dcur;
    float* dn = dnext;
    for (int it = 0; it < MAXIT; ++it) {
        // stage current distance vector (4 KB) into LDS (async-to-LDS path)
        stage_d_to_lds(dc, lds_d, t);
        __syncthreads();

        // wave w: sources [32w, 32w+32), lane l = local dst column
        float p = INF;
        const int sBase = w << 5;
#pragma unroll
        for (int k = 0; k < 32; ++k) {
            int s = sBase + k;
            p = fminf(p, lds_d[s] + lds_adj[s * DPB + l]);
        }
        lds_part[w * DPB + l] = p;
        __syncthreads();

        // fold the 32 per-wave partials; first 32 threads own one dst each
        if (t < DPB) {
            float c = INF;
#pragma unroll
            for (int w2 = 0; w2 < WPB; ++w2) c = fminf(c, lds_part[w2 * DPB + t]);
            const int dst  = dstBase + t;
            const float ov = lds_d[dst];
            const float nv = fminf(ov, c);
            dn[dst] = nv;
            if (nv < ov)
                __hip_atomic_fetch_or(&flags[it], 1, __ATOMIC_RELAXED,
                                      __HIP_MEMORY_SCOPE_AGENT);
        }

        grid_barrier(&bar[0], &bar[1]);

        // swap buffers; if nothing changed anywhere, the fixed point is reached
        float* tmp = dc; dc = dn; dn = tmp;
        int ch = __hip_atomic_load(&flags[it], __ATOMIC_ACQUIRE,
                                   __HIP_MEMORY_SCOPE_AGENT);
        if (ch == 0) break;          // identical for all blocks -> lockstep exit
        __syncthreads();
    }

    // ---- Epilogue: node_features = [embed | dist], diameter, eccentricity ----
    // One wave per node row: lanes copy the 256 embed floats, lane 0 appends dist.
    {
        const int row = blockIdx.x * WPB + w;            // 0..1023
        const float* erow = emb + (size_t)row * NF;
        float* orow = out + (size_t)row * OUTW;
        for (int j = l; j < NF; j += 32) orow[j] = erow[j];
        if (l == 0) orow[NF] = dc[row];
    }

    if (blockIdx.x == 0) {
        // max-reduce distances for the diameter; pick d[source] for eccentricity
        lds_part[t] = dc[t];
        __syncthreads();
        for (int sft = WGSIZE >> 1; sft > 0; sft >>= 1) {
            if (t < sft) lds_part[t] = fmaxf(lds_part[t], lds_part[t + sft]);
            __syncthreads();
        }
        if (t == 0) {
            out[(size_t)V * OUTW]     = lds_part[0];
            out[(size_t)V * OUTW + 1] = dc[*srcp];
        }
    }
}

// ---------------------------------------------------------------------------
// Host-side launch. ws layout: [d_cur(V) | d_next(V) | flags(V) | bar(2)] .
// ---------------------------------------------------------------------------
extern "C" void kernel_launch(void* const* d_in, const int* in_sizes, int n_in,
                              void* d_out, int out_size, void* d_ws, size_t ws_size,
                              hipStream_t stream) {
    (void)in_sizes; (void)n_in; (void)out_size; (void)ws_size;
    const float* adj  = (const float*)d_in[0];
    const float* emb  = (const float*)d_in[1];
    const int*   srcp = (const int*)d_in[2];
    float* out = (float*)d_out;

    float* dcur  = (float*)d_ws;
    float* dnext = dcur + V;
    int*   flags = (int*)(dnext + V);
    int*   bar   = flags + V;

    bf_init_kernel<<<1, V, 0, stream>>>(dcur, dnext, flags, bar, srcp);

    const size_t smem_bytes = (size_t)(V * DPB + V + WPB * DPB) * sizeof(float); // 136 KB
    bf_persistent_kernel<<<NBLK, WGSIZE, smem_bytes, stream>>>(
        adj, emb, srcp, out, dcur, dnext, flags, bar);
}